// DotProductPredictor_72232759984608
// MI455X (gfx1250) — compile-verified
//
#include <hip/hip_runtime.h>

// Per-edge dot product of gathered node features:
//   out[e] = dot(h[src[e]], h[dst[e]]),  h: [N,128] f32
//
// Roofline: 164 MFLOP vs ~655 MB of gathered row traffic -> pure data
// movement. h (51.2 MB) is L2-resident on MI455X (192 MB L2), so the gather
// runs at L2 bandwidth with only ~64 MB touching HBM (23.3 TB/s).
// Shape: one wave32 per edge; a 128-float row is exactly 32 lanes x float4,
// i.e. one fully coalesced 512 B burst (global_load_b128 per lane). Two edges
// in flight per wave for MLP; global_prefetch_b8 one iteration ahead hides
// the random-gather L2 hit latency. Reduction: 5x ds_swizzle XOR butterfly
// (one DS op per step, immediate pattern, no index setup). Wave-uniform index
// loads forced onto the scalar path via readfirstlane.
// WMMA is deliberately unused: diag(H_src . H_dst^T) wastes 15/16 of the
// matrix work (~2.6 GFLOP issued for 164 MFLOP useful) and would make this
// kernel compute-bound, i.e. slower than the memory floor.

#define D_FEAT 128

// ds_swizzle group-of-32 XOR pattern: offset = (xor_mask << 10) | and_mask(0x1f)
#define SWIZZLE_XOR(mask) (((mask) << 10) | 0x1f)

__device__ __forceinline__ float wave32_reduce_add(float v) {
    // EXEC is all-ones at every call site (all control flow is wave-uniform),
    // so ds_swizzle never reads a disabled lane.
    int x = __float_as_int(v);
    x = __float_as_int(__int_as_float(x) +
                       __int_as_float(__builtin_amdgcn_ds_swizzle(x, SWIZZLE_XOR(16))));
    x = __float_as_int(__int_as_float(x) +
                       __int_as_float(__builtin_amdgcn_ds_swizzle(x, SWIZZLE_XOR(8))));
    x = __float_as_int(__int_as_float(x) +
                       __int_as_float(__builtin_amdgcn_ds_swizzle(x, SWIZZLE_XOR(4))));
    x = __float_as_int(__int_as_float(x) +
                       __int_as_float(__builtin_amdgcn_ds_swizzle(x, SWIZZLE_XOR(2))));
    x = __float_as_int(__int_as_float(x) +
                       __int_as_float(__builtin_amdgcn_ds_swizzle(x, SWIZZLE_XOR(1))));
    return __int_as_float(x);
}

__global__ __launch_bounds__(256) void edge_dot_kernel(
    const float* __restrict__ h,
    const int* __restrict__ src,
    const int* __restrict__ dst,
    float* __restrict__ out,
    int n_edges)
{
    const int lane    = threadIdx.x & 31;                                 // wave32
    const int wave    = (int)((blockIdx.x * blockDim.x + threadIdx.x) >> 5);
    const int n_waves = (int)((gridDim.x * blockDim.x) >> 5);

    for (int e0v = wave; e0v < n_edges; e0v += 2 * n_waves) {
        // Wave-uniform edge ids -> scalar registers so index loads go through
        // the scalar/constant cache (s_load_b32), keeping the vector memory
        // pipe free for the 512 B row gathers.
        const int e0   = __builtin_amdgcn_readfirstlane(e0v);
        const int e1   = e0 + n_waves;
        const bool has1 = (e1 < n_edges);

        // ---- issue all gathers for this pair up front (4 x b128 per lane) ----
        const int s0 = src[e0];
        const int d0 = dst[e0];
        const float4 a0 = ((const float4*)(h + (size_t)s0 * D_FEAT))[lane];
        const float4 b0 = ((const float4*)(h + (size_t)d0 * D_FEAT))[lane];

        float4 a1 = {0.f, 0.f, 0.f, 0.f};
        float4 b1 = {0.f, 0.f, 0.f, 0.f};
        if (has1) {
            const int s1 = src[e1];
            const int d1 = dst[e1];
            a1 = ((const float4*)(h + (size_t)s1 * D_FEAT))[lane];
            b1 = ((const float4*)(h + (size_t)d1 * D_FEAT))[lane];
        }

        // ---- prefetch next pair's rows while we reduce this pair ----
        // Index stream is sequential (scalar-cache hits); the row gathers are
        // random, so global_prefetch_b8 pulls them toward the WGP early.
        const int ep0 = e0 + 2 * n_waves;
        const int ep1 = ep0 + n_waves;
        if (ep0 < n_edges) {
            const int sp = src[ep0];
            const int dp = dst[ep0];
            __builtin_prefetch(h + (size_t)sp * D_FEAT + lane * 4, 0, 3);
            __builtin_prefetch(h + (size_t)dp * D_FEAT + lane * 4, 0, 3);
        }
        if (ep1 < n_edges) {
            const int sp = src[ep1];
            const int dp = dst[ep1];
            __builtin_prefetch(h + (size_t)sp * D_FEAT + lane * 4, 0, 3);
            __builtin_prefetch(h + (size_t)dp * D_FEAT + lane * 4, 0, 3);
        }

        // ---- per-lane partial dots ----
        float p0 = fmaf(a0.x, b0.x, fmaf(a0.y, b0.y, fmaf(a0.z, b0.z, a0.w * b0.w)));
        float p1 = fmaf(a1.x, b1.x, fmaf(a1.y, b1.y, fmaf(a1.z, b1.z, a1.w * b1.w)));

        // ---- wave32 XOR-butterfly reductions (ds_swizzle, immediate pattern) ----
        p0 = wave32_reduce_add(p0);
        p1 = wave32_reduce_add(p1);

        if (lane == 0) {
            out[e0] = p0;
            if (has1) out[e1] = p1;
        }
    }
}

extern "C" void kernel_launch(void* const* d_in, const int* in_sizes, int n_in,
                              void* d_out, int out_size, void* d_ws, size_t ws_size,
                              hipStream_t stream) {
    const float* h   = (const float*)d_in[0];
    const int*   src = (const int*)d_in[1];   // reference int64 -> int32 under JAX default x64-off
    const int*   dst = (const int*)d_in[2];
    float*       out = (float*)d_out;
    const int n_edges = in_sizes[1];

    // 2048 blocks x 256 threads = 16384 wave32s; each wave grid-strides over
    // ~39 edges (2 per iteration) — ample parallelism to saturate L2/HBM.
    dim3 block(256);
    dim3 grid(2048);
    hipLaunchKernelGGL(edge_dot_kernel, grid, block, 0, stream,
                       h, src, dst, out, n_edges);
}